// GraphNeuralODE_32177894981950
// MI455X (gfx1250) — compile-verified
//
#include <hip/hip_runtime.h>
#include <cstdint>
#include <cstddef>

// Problem constants (from reference)
#define BB 8
#define NN 2000
#define TT 24
#define EG 64000
#define BN (BB * NN)            // 16000
#define E_ALL (BB * EG + BN)    // 528000
#define HOR 10

typedef float v2f __attribute__((ext_vector_type(2)));
typedef float v8f __attribute__((ext_vector_type(8)));

// ---------------------------------------------------------------------------
// Setup kernels (run once per launch)
// ---------------------------------------------------------------------------

__global__ void build_edges(const int64_t* __restrict__ ei,
                            int* __restrict__ src32, int* __restrict__ dst32,
                            int* __restrict__ deg) {
    int e = blockIdx.x * 256 + threadIdx.x;
    if (e >= E_ALL) return;
    int s, d;
    if (e < BB * EG) {
        int g = e / EG;
        int i = e - g * EG;
        s = (int)ei[i]      + g * NN;
        d = (int)ei[EG + i] + g * NN;
    } else {
        s = d = e - BB * EG;   // self loop
    }
    src32[e] = s;
    dst32[e] = d;
    atomicAdd(&deg[d], 1);
}

// single-block exclusive scan of 16000 degree counts -> row_start[16001]
__global__ void scan_deg(const int* __restrict__ deg, int* __restrict__ rs) {
    __shared__ int part[256];
    __shared__ int pref[257];
    const int CH = 63;                       // 256*63 = 16128 >= 16000
    int t = threadIdx.x;
    int base = t * CH;
    int s = 0;
    for (int j = 0; j < CH; ++j) {
        int i = base + j;
        if (i < BN) s += deg[i];
    }
    part[t] = s;
    __syncthreads();
    if (t == 0) {
        int a = 0;
        for (int i = 0; i < 256; ++i) { pref[i] = a; a += part[i]; }
        pref[256] = a;
    }
    __syncthreads();
    int a = pref[t];
    for (int j = 0; j < CH; ++j) {
        int i = base + j;
        if (i < BN) { rs[i] = a; a += deg[i]; }
    }
    if (t == 255) rs[BN] = pref[256];
}

__global__ void calc_dinv(const int* __restrict__ deg, float* __restrict__ dinv) {
    int i = blockIdx.x * 256 + threadIdx.x;
    if (i < BN) dinv[i] = rsqrtf((float)deg[i]);   // deg >= 1 (self loop)
}

__global__ void fill_csr(const int* __restrict__ src32, const int* __restrict__ dst32,
                         const int* __restrict__ rs, int* __restrict__ cursor,
                         const float* __restrict__ dinv,
                         int* __restrict__ csrc, float* __restrict__ cnorm) {
    int e = blockIdx.x * 256 + threadIdx.x;
    if (e >= E_ALL) return;
    int s = src32[e], d = dst32[e];
    int p = atomicAdd(&cursor[d], 1);
    int idx = rs[d] + p;
    csrc[idx] = s;
    cnorm[idx] = dinv[s] * dinv[d];
}

__global__ void init_y(const float* __restrict__ x, float* __restrict__ y,
                       float* __restrict__ out) {
    int i = blockIdx.x * 256 + threadIdx.x;
    if (i < BN) {
        float v = x[i * TT + (TT - 1)];
        y[i] = v;
        out[i * HOR] = v;       // preds at h = 0
    }
}

// ---------------------------------------------------------------------------
// GCN layer kernels
// ---------------------------------------------------------------------------

// scalar aggregation: vout[n] = sum_e vin[src]*norm (+ bias[0] if given)
__global__ void agg_scalar(const float* __restrict__ vin,
                           const int* __restrict__ rs, const int* __restrict__ csrc,
                           const float* __restrict__ cnorm,
                           float* __restrict__ vout, const float* __restrict__ bias) {
    int n = blockIdx.x * 256 + threadIdx.x;
    if (n >= BN) return;
    float acc = 0.f;
    int s = rs[n], e = rs[n + 1];
    for (int i = s; i < e; ++i)
        acc = fmaf(vin[csrc[i]], cnorm[i], acc);
    if (bias) acc += bias[0];
    vout[n] = acc;
}

// layer-0 expansion: H[n][f] = tanh(z[n]*W0[f] + b0[f])
__global__ void expand_l0(const float* __restrict__ z, const float* __restrict__ W0,
                          const float* __restrict__ b0, float* __restrict__ H) {
    int idx = blockIdx.x * 256 + threadIdx.x;   // BN*64 threads
    int n = idx >> 6, f = idx & 63;
    H[idx] = tanhf(fmaf(z[n], W0[f], b0[f]));
}

// 64-wide aggregation: one wave per node, float2 per lane (coalesced 256B gathers)
__global__ __launch_bounds__(256) void agg64(const float2* __restrict__ Hin,
                                             const int* __restrict__ rs,
                                             const int* __restrict__ csrc,
                                             const float* __restrict__ cnorm,
                                             const float* __restrict__ bias,
                                             float2* __restrict__ Hout,
                                             int applyTanh) {
    int node = blockIdx.x * 8 + (threadIdx.x >> 5);
    int lane = threadIdx.x & 31;
    float2 acc = make_float2(0.f, 0.f);
    int s = rs[node], e = rs[node + 1];
    for (int i = s; i < e; ++i) {
        int src = csrc[i];
        float w = cnorm[i];
        float2 h = Hin[src * 32 + lane];
        acc.x = fmaf(h.x, w, acc.x);
        acc.y = fmaf(h.y, w, acc.y);
    }
    const float2 bv = ((const float2*)bias)[lane];
    acc.x += bv.x;
    acc.y += bv.y;
    if (applyTanh) { acc.x = tanhf(acc.x); acc.y = tanhf(acc.y); }
    Hout[node * 32 + lane] = acc;
}

// fp32 WMMA GEMM: H(16000x64) = X(16000x64) @ W(64x64)
// block = 256 threads = 8 waves; each wave computes a 16x64 row stripe.
__global__ __launch_bounds__(256) void gemm64_wmma(const float* __restrict__ X,
                                                   const float* __restrict__ W,
                                                   float* __restrict__ Hout) {
    __shared__ float Wl[64 * 64];
    __shared__ float Xl[128 * 65];      // +1 pad to avoid bank conflicts
    const int tid = threadIdx.x;
    const int rowBase = blockIdx.x * 128;

    // stage W (16KB) into LDS
    {
        const float4* Wg = (const float4*)W;
        float4* Wl4 = (float4*)Wl;
        #pragma unroll
        for (int i = 0; i < 4; ++i) Wl4[tid + 256 * i] = Wg[tid + 256 * i];
    }
    // stage 128x64 X tile into LDS (padded stride 65)
    #pragma unroll
    for (int i = 0; i < 8; ++i) {
        int idx = tid + 256 * i;                 // float4 index 0..2047
        int r = idx >> 4;                        // 16 float4 per row
        int c = (idx & 15) << 2;
        float4 val = *(const float4*)(X + (size_t)(rowBase + r) * 64 + c);
        float* dst = &Xl[r * 65 + c];
        dst[0] = val.x; dst[1] = val.y; dst[2] = val.z; dst[3] = val.w;
    }
    __syncthreads();

    const int wv = tid >> 5;
    const int lane = tid & 31;
    const int rl = (wv << 4) + (lane & 15);     // local row for A fragment
    const int kOff = (lane >> 4) << 1;          // K sub-offset per A/B layout
    const int n = lane & 15;

    v8f acc0 = {}, acc1 = {}, acc2 = {}, acc3 = {};
    #pragma unroll
    for (int kc = 0; kc < 16; ++kc) {
        const int k = 4 * kc + kOff;
        v2f a;
        a.x = Xl[rl * 65 + k];
        a.y = Xl[rl * 65 + k + 1];
        v2f b0, b1, b2, b3;
        b0.x = Wl[k * 64 + n];        b0.y = Wl[(k + 1) * 64 + n];
        b1.x = Wl[k * 64 + 16 + n];   b1.y = Wl[(k + 1) * 64 + 16 + n];
        b2.x = Wl[k * 64 + 32 + n];   b2.y = Wl[(k + 1) * 64 + 32 + n];
        b3.x = Wl[k * 64 + 48 + n];   b3.y = Wl[(k + 1) * 64 + 48 + n];
        acc0 = __builtin_amdgcn_wmma_f32_16x16x4_f32(false, a, false, b0, (short)0, acc0, false, false);
        acc1 = __builtin_amdgcn_wmma_f32_16x16x4_f32(false, a, false, b1, (short)0, acc1, false, false);
        acc2 = __builtin_amdgcn_wmma_f32_16x16x4_f32(false, a, false, b2, (short)0, acc2, false, false);
        acc3 = __builtin_amdgcn_wmma_f32_16x16x4_f32(false, a, false, b3, (short)0, acc3, false, false);
    }

    // C/D layout: VGPR v -> M = v (lanes 0-15) / v+8 (lanes 16-31); N = lane&15
    const int rowW = rowBase + (wv << 4) + ((lane >> 4) << 3);
    const int col = lane & 15;
    #pragma unroll
    for (int v = 0; v < 8; ++v) {
        float* o = Hout + (size_t)(rowW + v) * 64 + col;
        o[0]  = acc0[v];
        o[16] = acc1[v];
        o[32] = acc2[v];
        o[48] = acc3[v];
    }
}

// per-node dot with W4 column: z[n] = dot(H[n][:], W4)
__global__ __launch_bounds__(256) void dot_w4(const float* __restrict__ H,
                                              const float* __restrict__ W4,
                                              float* __restrict__ z) {
    int wv = (blockIdx.x * 256 + threadIdx.x) >> 5;   // global wave = node
    int lane = threadIdx.x & 31;
    float a = fmaf(H[wv * 64 + lane], W4[lane],
                   H[wv * 64 + 32 + lane] * W4[32 + lane]);
    #pragma unroll
    for (int off = 16; off > 0; off >>= 1)
        a += __shfl_down(a, off, 32);
    if (lane == 0) z[wv] = a;
}

// RK4 combination: outv = y + c1*k1 + c2*k2 + c3*k3 + c4*k4 (+ preds write)
__global__ void combine(const float* __restrict__ y,
                        const float* __restrict__ k1, const float* __restrict__ k2,
                        const float* __restrict__ k3, const float* __restrict__ k4,
                        float c1, float c2, float c3, float c4,
                        float* __restrict__ outv, float* __restrict__ preds, int h) {
    int i = blockIdx.x * 256 + threadIdx.x;
    if (i >= BN) return;
    float v = y[i];
    v = fmaf(c1, k1[i], v);
    v = fmaf(c2, k2[i], v);
    v = fmaf(c3, k3[i], v);
    v = fmaf(c4, k4[i], v);
    outv[i] = v;
    if (preds) preds[i * HOR + h] = v;
}

// ---------------------------------------------------------------------------
// Host launch
// ---------------------------------------------------------------------------

extern "C" void kernel_launch(void* const* d_in, const int* in_sizes, int n_in,
                              void* d_out, int out_size, void* d_ws, size_t ws_size,
                              hipStream_t stream) {
    (void)in_sizes; (void)n_in; (void)out_size; (void)ws_size;

    const float*   x  = (const float*)d_in[0];
    const int64_t* ei = (const int64_t*)d_in[1];
    const float *W0 = (const float*)d_in[2],  *b0 = (const float*)d_in[3];
    const float *W1 = (const float*)d_in[4],  *b1 = (const float*)d_in[5];
    const float *W2 = (const float*)d_in[6],  *b2 = (const float*)d_in[7];
    const float *W3 = (const float*)d_in[8],  *b3 = (const float*)d_in[9];
    const float *W4 = (const float*)d_in[10], *b4 = (const float*)d_in[11];
    float* out = (float*)d_out;

    // workspace carve-up (256B aligned chunks)
    char* p = (char*)d_ws;
    auto carve = [&](size_t bytes) -> void* {
        void* r = (void*)p;
        p += (bytes + 255) & ~(size_t)255;
        return r;
    };
    int*   src32  = (int*)  carve(sizeof(int)   * E_ALL);
    int*   dst32  = (int*)  carve(sizeof(int)   * E_ALL);
    int*   deg    = (int*)  carve(sizeof(int)   * BN);
    int*   rs     = (int*)  carve(sizeof(int)   * (BN + 1));
    int*   cursor = (int*)  carve(sizeof(int)   * BN);
    int*   csrc   = (int*)  carve(sizeof(int)   * E_ALL);
    float* cnorm  = (float*)carve(sizeof(float) * E_ALL);
    float* dinv   = (float*)carve(sizeof(float) * BN);
    float* y      = (float*)carve(sizeof(float) * BN);
    float* k1     = (float*)carve(sizeof(float) * BN);
    float* k2     = (float*)carve(sizeof(float) * BN);
    float* k3     = (float*)carve(sizeof(float) * BN);
    float* k4     = (float*)carve(sizeof(float) * BN);
    float* ytmp   = (float*)carve(sizeof(float) * BN);
    float* z      = (float*)carve(sizeof(float) * BN);
    float* hA     = (float*)carve(sizeof(float) * BN * 64);
    float* hB     = (float*)carve(sizeof(float) * BN * 64);

    const int GE = (E_ALL + 255) / 256;   // edge-grid
    const int GN = (BN + 255) / 256;      // node-grid (63)

    // ---- build normalized CSR once ----
    hipMemsetAsync(deg,    0, sizeof(int) * BN, stream);
    hipMemsetAsync(cursor, 0, sizeof(int) * BN, stream);
    build_edges<<<GE, 256, 0, stream>>>(ei, src32, dst32, deg);
    scan_deg<<<1, 256, 0, stream>>>(deg, rs);
    calc_dinv<<<GN, 256, 0, stream>>>(deg, dinv);
    fill_csr<<<GE, 256, 0, stream>>>(src32, dst32, rs, cursor, dinv, csrc, cnorm);

    // ---- initial state + preds[:, :, 0] ----
    init_y<<<GN, 256, 0, stream>>>(x, y, out);

    // ---- 5-layer GCN ODE function ----
    auto eval_f = [&](const float* yin, float* kout) {
        // L0: aggregate scalar, then expand through W0 (+b0, tanh)
        agg_scalar<<<GN, 256, 0, stream>>>(yin, rs, csrc, cnorm, z, nullptr);
        expand_l0<<<(BN * 64) / 256, 256, 0, stream>>>(z, W0, b0, hA);
        // L1..L3: fp32 WMMA GEMM + 64-wide aggregation (+bias, tanh)
        gemm64_wmma<<<BN / 128, 256, 0, stream>>>(hA, W1, hB);
        agg64<<<BN / 8, 256, 0, stream>>>((const float2*)hB, rs, csrc, cnorm, b1, (float2*)hA, 1);
        gemm64_wmma<<<BN / 128, 256, 0, stream>>>(hA, W2, hB);
        agg64<<<BN / 8, 256, 0, stream>>>((const float2*)hB, rs, csrc, cnorm, b2, (float2*)hA, 1);
        gemm64_wmma<<<BN / 128, 256, 0, stream>>>(hA, W3, hB);
        agg64<<<BN / 8, 256, 0, stream>>>((const float2*)hB, rs, csrc, cnorm, b3, (float2*)hA, 1);
        // L4: per-node dot with W4, then scalar aggregation + b4 (no tanh)
        dot_w4<<<BN / 8 / 8 * 8 / 8, 256, 0, stream>>>(hA, W4, z); // 250 blocks x 8 = 2000 waves (nodes 0..1999)
        agg_scalar<<<GN, 256, 0, stream>>>(z, rs, csrc, cnorm, kout, b4);
    };

    // ---- RK4 (3/8 rule), 9 steps ----
    const float dt = (float)HOR / (float)(HOR - 1);
    for (int step = 0; step < HOR - 1; ++step) {
        eval_f(y, k1);
        combine<<<GN, 256, 0, stream>>>(y, k1, k1, k1, k1,
                                        dt / 3.f, 0.f, 0.f, 0.f, ytmp, nullptr, 0);
        eval_f(ytmp, k2);
        combine<<<GN, 256, 0, stream>>>(y, k1, k2, k2, k2,
                                        -dt / 3.f, dt, 0.f, 0.f, ytmp, nullptr, 0);
        eval_f(ytmp, k3);
        combine<<<GN, 256, 0, stream>>>(y, k1, k2, k3, k3,
                                        dt, -dt, dt, 0.f, ytmp, nullptr, 0);
        eval_f(ytmp, k4);
        combine<<<GN, 256, 0, stream>>>(y, k1, k2, k3, k4,
                                        dt * 0.125f, dt * 0.375f, dt * 0.375f, dt * 0.125f,
                                        y, out, step + 1);
    }
}